// GraphGNN_54065048322533
// MI455X (gfx1250) — compile-verified
//
#include <hip/hip_runtime.h>
#include <hip/hip_bf16.h>
#include <math.h>

#define B_ 16
#define N_ 2000
#define E_ 32000
#define F_ 8
#define H_ 16
#define M_ 32
#define BE (B_*E_)

typedef float v2f __attribute__((ext_vector_type(2)));
typedef float v8f __attribute__((ext_vector_type(8)));

// ---------------- kernel 0: zero the fixed-point row accumulators ----------
__global__ __launch_bounds__(256) void zero_acc_kernel(unsigned long long* __restrict__ acc) {
    int i = blockIdx.x * 256 + threadIdx.x;
    if (i < B_ * N_) acc[i] = 0ull;
}

// ---------------- kernel 1: edge_attr mean / inv-std (ddof=1) --------------
__global__ __launch_bounds__(512) void stats_kernel(const float* __restrict__ ea,
                                                    float* __restrict__ stats) {
    __shared__ double sh[4][512];
    int tid = threadIdx.x;
    double s0 = 0, s1 = 0, q0 = 0, q1 = 0;
    for (int i = tid; i < E_; i += 512) {
        double d0 = (double)ea[2 * i + 0];
        double d1 = (double)ea[2 * i + 1];
        s0 += d0; s1 += d1; q0 += d0 * d0; q1 += d1 * d1;
    }
    sh[0][tid] = s0; sh[1][tid] = s1; sh[2][tid] = q0; sh[3][tid] = q1;
    __syncthreads();
    for (int off = 256; off > 0; off >>= 1) {
        if (tid < off) {
            sh[0][tid] += sh[0][tid + off];
            sh[1][tid] += sh[1][tid + off];
            sh[2][tid] += sh[2][tid + off];
            sh[3][tid] += sh[3][tid + off];
        }
        __syncthreads();
    }
    if (tid == 0) {
        double m0 = sh[0][0] / E_, m1 = sh[1][0] / E_;
        double v0 = (sh[2][0] - (double)E_ * m0 * m0) / (double)(E_ - 1);
        double v1 = (sh[3][0] - (double)E_ * m1 * m1) / (double)(E_ - 1);
        stats[0] = (float)m0;
        stats[1] = (float)m1;
        stats[2] = (float)(1.0 / sqrt(v0));
        stats[3] = (float)(1.0 / sqrt(v1));
    }
}

// ---------------- kernel 2: per-edge GRU + WMMA MLP + row-sum atomics ------
// 256 threads = 8 waves; each wave handles 32 edges (1 GRU per lane, then
// two 16-edge WMMA tiles: D(16x16) = A(16x4 f32) x B(4x16 f32), K=16 chained).
__global__ __launch_bounds__(256) void edge_kernel(
    const float* __restrict__ x, const float* __restrict__ edge_attr,
    const float* __restrict__ wind_mean, const float* __restrict__ wind_std,
    const float* __restrict__ e0,
    const float* __restrict__ W_ih, const float* __restrict__ b_ih,
    const float* __restrict__ W_hh, const float* __restrict__ b_hh,
    const float* __restrict__ W1, const float* __restrict__ b1,
    const float* __restrict__ W2, const float* __restrict__ b2,
    const int* __restrict__ edge_index, const float* __restrict__ stats,
    unsigned long long* __restrict__ rowacc, float* __restrict__ e_exp) {

    __shared__ float sW_ih[48 * 3];
    __shared__ float sb_ih[48];
    __shared__ float sW_hh[48 * 16];
    __shared__ float sb_hh[48];
    __shared__ float hbuf[8][32 * 17];   // per-wave: 32 edges x 16 h (+pad)

    const int tid = threadIdx.x;
    for (int i = tid; i < 144; i += 256) sW_ih[i] = W_ih[i];
    for (int i = tid; i < 48;  i += 256) { sb_ih[i] = b_ih[i]; sb_hh[i] = b_hh[i]; }
    for (int i = tid; i < 768; i += 256) sW_hh[i] = W_hh[i];
    __syncthreads();

    const int wave = tid >> 5;
    const int lane = tid & 31;
    const int wave_base = (blockIdx.x * 8 + wave) * 32;   // first global edge id of wave
    const int ge = wave_base + lane;                      // this lane's edge (always < BE)
    const int b  = ge / E_;
    const int e  = ge - b * E_;

    // ---- per-lane GRU for one edge ----
    const int   src   = edge_index[e];
    const float speed = x[(b * N_ + src) * F_ + 6] * wind_std[0] + wind_mean[0];
    const float direc = x[(b * N_ + src) * F_ + 7] * wind_std[1] + wind_mean[1];
    const float cdist = edge_attr[2 * e + 0];
    const float cdir  = edge_attr[2 * e + 1];
    const float theta = fabsf(cdir - direc);
    const float wgt   = fmaxf(0.0f, 3.0f * speed * cosf(theta) / cdist);
    const float a0 = (cdist - stats[0]) * stats[2];
    const float a1 = (cdir  - stats[1]) * stats[3];

    float hp[16];
    const float* e0p = e0 + (size_t)ge * 16;
    #pragma unroll
    for (int j = 0; j < 16; j++) hp[j] = e0p[j];

    float* hout = &hbuf[wave][lane * 17];
    #pragma unroll
    for (int k = 0; k < 16; k++) {
        float gi_r = sW_ih[k*3+0]*a0      + sW_ih[k*3+1]*a1      + sW_ih[k*3+2]*wgt      + sb_ih[k];
        float gi_z = sW_ih[(16+k)*3+0]*a0 + sW_ih[(16+k)*3+1]*a1 + sW_ih[(16+k)*3+2]*wgt + sb_ih[16+k];
        float gi_n = sW_ih[(32+k)*3+0]*a0 + sW_ih[(32+k)*3+1]*a1 + sW_ih[(32+k)*3+2]*wgt + sb_ih[32+k];
        float gh_r = sb_hh[k], gh_z = sb_hh[16+k], gh_n = sb_hh[32+k];
        #pragma unroll
        for (int j = 0; j < 16; j++) {
            gh_r += sW_hh[k*16+j]      * hp[j];
            gh_z += sW_hh[(16+k)*16+j] * hp[j];
            gh_n += sW_hh[(32+k)*16+j] * hp[j];
        }
        float rg = 1.0f / (1.0f + expf(-(gi_r + gh_r)));
        float zg = 1.0f / (1.0f + expf(-(gi_z + gh_z)));
        float ng = tanhf(gi_n + rg * gh_n);
        hout[k] = ng + zg * (hp[k] - ng);
    }
    __syncthreads();

    // ---- WMMA MLP: (32 edges x 16) @ W1^T(16x32), K chained via 16x16x4 f32 ----
    const int ln    = lane & 15;
    const int half  = lane >> 4;
    const int khalf = half * 2;   // A/B: lanes 0-15 carry K=4s+0/1, lanes 16-31 K=4s+2/3

    v2f bfrag[2][4];              // [output tile][K step]  from W1 (M_=32 x H_=16 row-major)
    #pragma unroll
    for (int t = 0; t < 2; t++) {
        #pragma unroll
        for (int s = 0; s < 4; s++) {
            int k = 4 * s + khalf;
            int n = t * 16 + ln;
            bfrag[t][s][0] = W1[n * 16 + k];
            bfrag[t][s][1] = W1[n * 16 + k + 1];
        }
    }

    const float b1v0 = b1[ln],  b1v1 = b1[16 + ln];
    const float w2v0 = W2[ln],  w2v1 = W2[16 + ln];
    const float b2v  = b2[0];
    const float* hb  = &hbuf[wave][0];

    #pragma unroll
    for (int et = 0; et < 2; et++) {
        v8f acc0 = {}; v8f acc1 = {};
        const int eidx = 16 * et + ln;
        #pragma unroll
        for (int s = 0; s < 4; s++) {
            v2f a;
            int k = 4 * s + khalf;
            a[0] = hb[eidx * 17 + k];
            a[1] = hb[eidx * 17 + k + 1];
            acc0 = __builtin_amdgcn_wmma_f32_16x16x4_f32(false, a, false, bfrag[0][s],
                                                         (short)0, acc0, false, false);
            acc1 = __builtin_amdgcn_wmma_f32_16x16x4_f32(false, a, false, bfrag[1][s],
                                                         (short)0, acc1, false, false);
        }
        // epilogue: relu + W2 dot (cross-lane reduce over 16 lanes per half)
        #pragma unroll
        for (int r = 0; r < 8; r++) {
            float p = fmaxf(acc0[r] + b1v0, 0.0f) * w2v0 +
                      fmaxf(acc1[r] + b1v1, 0.0f) * w2v1;
            p += __shfl_xor(p, 1, 32);
            p += __shfl_xor(p, 2, 32);
            p += __shfl_xor(p, 4, 32);
            p += __shfl_xor(p, 8, 32);
            if (ln == 0) {   // lanes 0 & 16: edge m = 16*et + r + 8*half
                int geid = wave_base + 16 * et + r + 8 * half;
                float y  = fmaxf(p + b2v, 0.0f);
                float ex = expf(y);
                e_exp[geid] = ex;
                int bb = geid / E_;
                int ee = geid - bb * E_;
                int sv = edge_index[ee];
                unsigned long long fx =
                    (unsigned long long)(((double)ex - 1.0) * 4294967296.0);
                atomicAdd(&rowacc[bb * N_ + sv], fx);
            }
        }
    }
}

// ---------------- kernel 3: finalize rows -> 1/(N + sum) -------------------
__global__ __launch_bounds__(256) void finalize_kernel(const unsigned long long* __restrict__ acc,
                                                       float* __restrict__ inv) {
    int i = blockIdx.x * 256 + threadIdx.x;
    if (i < B_ * N_) {
        double s = (double)N_ + (double)acc[i] * (1.0 / 4294967296.0);
        inv[i] = (float)(1.0 / s);
    }
}

// ---------------- kernel 4: fill dense output with 1/rowsum (float4) -------
__global__ __launch_bounds__(256) void fill_kernel(const float* __restrict__ inv,
                                                   float* __restrict__ out) {
    int tid = blockIdx.x * 256 + threadIdx.x;     // B_*N_*(N_/4) = 16,000,000
    if (tid >= B_ * N_ * (N_ / 4)) return;
    int row = tid / (N_ / 4);
    int q   = tid - row * (N_ / 4);
    float v = inv[row];
    float4 f = make_float4(v, v, v, v);
    *(float4*)(out + (size_t)row * N_ + (size_t)q * 4) = f;
}

// ---------------- kernel 5: scatter edge softmax values --------------------
__global__ __launch_bounds__(256) void scatter_kernel(const int* __restrict__ ei,
                                                      const float* __restrict__ e_exp,
                                                      const float* __restrict__ inv,
                                                      float* __restrict__ out) {
    int g = blockIdx.x * 256 + threadIdx.x;
    if (g >= BE) return;
    int b = g / E_;
    int e = g - b * E_;
    int s = ei[e];
    int t = ei[E_ + e];
    out[(size_t)b * N_ * N_ + (size_t)s * N_ + t] = e_exp[g] * inv[b * N_ + s];
}

extern "C" void kernel_launch(void* const* d_in, const int* in_sizes, int n_in,
                              void* d_out, int out_size, void* d_ws, size_t ws_size,
                              hipStream_t stream) {
    const float* x         = (const float*)d_in[0];
    const float* edge_attr = (const float*)d_in[1];
    const float* wind_mean = (const float*)d_in[2];
    const float* wind_std  = (const float*)d_in[3];
    const float* e0        = (const float*)d_in[4];
    const float* W_ih      = (const float*)d_in[5];
    const float* b_ih      = (const float*)d_in[6];
    const float* W_hh      = (const float*)d_in[7];
    const float* b_hh      = (const float*)d_in[8];
    const float* W1        = (const float*)d_in[9];
    const float* b1        = (const float*)d_in[10];
    const float* W2        = (const float*)d_in[11];
    const float* b2        = (const float*)d_in[12];
    const int*   ei        = (const int*)d_in[13];

    float* out = (float*)d_out;

    // workspace layout
    char* ws = (char*)d_ws;
    float*              stats  = (float*)(ws + 0);                 // 4 floats
    unsigned long long* rowacc = (unsigned long long*)(ws + 256);  // 32000 * 8B
    float*              inv    = (float*)(ws + 262144);            // 32000 * 4B
    float*              e_exp  = (float*)(ws + 393216);            // 512000 * 4B

    zero_acc_kernel<<<(B_ * N_ + 255) / 256, 256, 0, stream>>>(rowacc);
    stats_kernel<<<1, 512, 0, stream>>>(edge_attr, stats);
    edge_kernel<<<BE / 256, 256, 0, stream>>>(x, edge_attr, wind_mean, wind_std, e0,
                                              W_ih, b_ih, W_hh, b_hh, W1, b1, W2, b2,
                                              ei, stats, rowacc, e_exp);
    finalize_kernel<<<(B_ * N_ + 255) / 256, 256, 0, stream>>>(rowacc, inv);
    fill_kernel<<<(B_ * N_ * (N_ / 4) + 255) / 256, 256, 0, stream>>>(inv, out);
    scatter_kernel<<<(BE + 255) / 256, 256, 0, stream>>>(ei, e_exp, inv, out);
}